// weighted_GCN_89687507076053
// MI455X (gfx1250) — compile-verified
//
#include <hip/hip_runtime.h>

// Problem constants (match reference)
#define NN    50000
#define EE    800000
#define TT    4
#define FF    64
#define MROWS (NN * TT)          // 200000 rows for the (N*T, F) GEMMs
#define SBIG  (NN * TT * FF)     // 12,800,000 elements per big tensor
#define BNEPS 1e-5f

typedef __attribute__((ext_vector_type(2))) float v2f;
typedef __attribute__((ext_vector_type(8))) float v8f;

// ---------------------------------------------------------------------------
// Zero fill (float4-vectorized, grid-stride) — graph-capture-safe memset
// ---------------------------------------------------------------------------
__global__ void zero_f4(float4* __restrict__ p, int n4) {
  int i = blockIdx.x * blockDim.x + threadIdx.x;
  int stride = gridDim.x * blockDim.x;
  float4 z = make_float4(0.f, 0.f, 0.f, 0.f);
  for (; i < n4; i += stride) p[i] = z;
}

// ---------------------------------------------------------------------------
// h0[n,t,f] = nf[n,t,f] + pe[ids[t], f]   (float4 over f; F=64, T=4 pow2)
// ---------------------------------------------------------------------------
__global__ void add_pe(const float4* __restrict__ nf, const float4* __restrict__ pe4,
                       const int* __restrict__ ids, float4* __restrict__ h0) {
  int i = blockIdx.x * blockDim.x + threadIdx.x;   // over SBIG/4, exact grid
  if (i >= SBIG / 4) return;
  int f4 = i & (FF / 4 - 1);           // float4 index within row (16 per row)
  int t  = (i >> 4) & (TT - 1);        // row = i/16 = n*T + t
  float4 v = nf[i];
  float4 p = pe4[ids[t] * (FF / 4) + f4];
  v.x += p.x; v.y += p.y; v.z += p.z; v.w += p.w;
  h0[i] = v;
}

// ---------------------------------------------------------------------------
// Gather-mul-scatter: agg[dst_e, t, f] += h[src_e, t, f] * ew[t, e]
// One 256-thread block per edge: tid -> (t = tid/64, f = tid%64).
// agg (51.2 MB) is L2-resident on MI455X (192 MB L2) -> atomics stay in L2.
// ---------------------------------------------------------------------------
__global__ void scatter_edges(const float* __restrict__ h, const float* __restrict__ ew,
                              const int* __restrict__ src, const int* __restrict__ dst,
                              float* __restrict__ agg) {
  int e = blockIdx.x;
  int t = threadIdx.x >> 6;
  int f = threadIdx.x & 63;
  int s = src[e];
  int d = dst[e];
  float w = ew[t * EE + e];
  float v = h[(s * TT + t) * FF + f] * w;
  atomicAdd(&agg[(d * TT + t) * FF + f], v);
}

// ---------------------------------------------------------------------------
// X = A @ W, A: (MROWS x 64) f32, W: (64 x 64) f32 row-major.
// One wave per 16x16 output tile; 16 steps of V_WMMA_F32_16X16X4_F32.
// Fragment layouts per CDNA5 ISA 7.12.2 (f32):
//   A 16x4 : lanes 0-15 row=lane  {K0,K1}; lanes 16-31 row=lane-16 {K2,K3}
//   B 4x16 : lanes 0-15 col=lane  {K0,K1}; lanes 16-31 col=lane-16 {K2,K3}
//   C/D    : vgpr i -> row m0 + 8*half + i, col n0 + (lane&15)
// ---------------------------------------------------------------------------
__global__ void __launch_bounds__(256)
gemm_wmma_f32(const float* __restrict__ A, const float* __restrict__ W,
              float* __restrict__ X) {
  int gwave = (blockIdx.x * blockDim.x + threadIdx.x) >> 5;
  int lane  = threadIdx.x & 31;
  int nt = gwave & 3;                    // 4 column tiles (64 cols)
  int mt = gwave >> 2;                   // 12500 row tiles
  if (mt >= MROWS / 16) return;          // wave-uniform (EXEC stays all-1s)
  int m0 = mt * 16, n0 = nt * 16;
  int half = lane >> 4;
  int lr   = lane & 15;

  const float* Arow = A + (m0 + lr) * FF;    // 256B-aligned row
  const float* Wc   = W + n0 + lr;
  v8f c = {};
#pragma unroll
  for (int kk = 0; kk < 16; ++kk) {
    int k0 = kk * 4 + half * 2;              // even -> 8B-aligned v2f load
    v2f a = *(const v2f*)(Arow + k0);
    v2f b;
    b.x = Wc[k0 * FF];
    b.y = Wc[(k0 + 1) * FF];
    c = __builtin_amdgcn_wmma_f32_16x16x4_f32(
        /*neg_a=*/false, a, /*neg_b=*/false, b,
        /*c_mod=*/(short)0, c, /*reuse_a=*/false, /*reuse_b=*/false);
  }
  float* Xc = X + (m0 + half * 8) * FF + n0 + lr;
#pragma unroll
  for (int i = 0; i < 8; ++i) Xc[i * FF] = c[i];
}

// ---------------------------------------------------------------------------
// BN stats: stats[f] = sum over MROWS rows, stats[64+f] = sum of squares
// ---------------------------------------------------------------------------
__global__ void bn_stats(const float* __restrict__ X, float* __restrict__ stats) {
  __shared__ float s[128];
  int tid = threadIdx.x;                 // 256 threads
  if (tid < 128) s[tid] = 0.f;
  __syncthreads();
  int f  = tid & 63;
  int rg = tid >> 6;                     // 4 row-groups per block
  float sum = 0.f, sq = 0.f;
  for (int row = blockIdx.x * 4 + rg; row < MROWS; row += gridDim.x * 4) {
    float v = X[row * FF + f];
    sum += v; sq += v * v;
  }
  atomicAdd(&s[f], sum);
  atomicAdd(&s[64 + f], sq);
  __syncthreads();
  if (tid < 128) atomicAdd(&stats[tid], s[tid]);
}

// ---------------------------------------------------------------------------
// Y = relu((X - mean) * rsqrt(var + eps) * gamma + beta), per-channel f
// ---------------------------------------------------------------------------
__global__ void bn_relu(const float* __restrict__ X, const float* __restrict__ stats,
                        const float* __restrict__ gamma, const float* __restrict__ beta,
                        float* __restrict__ Y) {
  int i = blockIdx.x * blockDim.x + threadIdx.x;   // exact grid over SBIG
  if (i >= SBIG) return;
  int f = i & 63;
  const float invM = 1.0f / (float)MROWS;
  float mean = stats[f] * invM;
  float var  = stats[64 + f] * invM - mean * mean;
  float inv  = rsqrtf(var + BNEPS);
  float y = (X[i] - mean) * inv * gamma[f] + beta[f];
  Y[i] = fmaxf(y, 0.f);
}

// ---------------------------------------------------------------------------
// Column sums of agg over nodes: cs[c] = sum_n agg[n, c], c in [0, T*F)
// ---------------------------------------------------------------------------
__global__ void col_sum(const float* __restrict__ A, float* __restrict__ cs) {
  int c = threadIdx.x;                   // 256 = T*F channels
  float sum = 0.f;
  for (int n = blockIdx.x; n < NN; n += gridDim.x)
    sum += A[n * (TT * FF) + c];
  atomicAdd(&cs[c], sum);
}

// readout[t, n] = (colsum[t, :] / N) @ W[:, n]   — tiny 4x64x64, one block
__global__ void readout_k(const float* __restrict__ cs, const float* __restrict__ W,
                          float* __restrict__ out) {
  int tid = threadIdx.x;                 // 256
  int t = tid >> 6, n = tid & 63;
  float acc = 0.f;
#pragma unroll
  for (int k = 0; k < 64; ++k)
    acc += cs[t * 64 + k] * W[k * 64 + n];
  out[tid] = acc * (1.0f / (float)NN);
}

// ---------------------------------------------------------------------------
extern "C" void kernel_launch(void* const* d_in, const int* in_sizes, int n_in,
                              void* d_out, int out_size, void* d_ws, size_t ws_size,
                              hipStream_t stream) {
  const float* nf  = (const float*)d_in[0];   // (N,T,F)
  const float* ew  = (const float*)d_in[1];   // (T,E)
  const int*   ids = (const int*)d_in[2];     // (T,)
  const int*   esr = (const int*)d_in[3];     // (E,)
  const int*   eds = (const int*)d_in[4];     // (E,)
  const float* pe  = (const float*)d_in[5];   // (MAX_LEN,F)
  const float* W0  = (const float*)d_in[6];
  const float* W1  = (const float*)d_in[7];
  const float* g0  = (const float*)d_in[8];
  const float* b0  = (const float*)d_in[9];
  const float* g1  = (const float*)d_in[10];
  const float* b1  = (const float*)d_in[11];
  float* out = (float*)d_out;                 // [SBIG h] ++ [256 readout]

  // Workspace: two 51.2MB ping-pong buffers + 512 small floats
  float* bufA   = (float*)d_ws;               // h0 -> x0 -> agg1
  float* bufB   = bufA + SBIG;                // agg0 -> h1 -> x1
  float* stats0 = bufB + SBIG;                // 128
  float* stats1 = stats0 + 128;               // 128
  float* csum   = stats1 + 128;               // 256

  const int TPB = 256;
  dim3 blkElem4((SBIG / 4 + TPB - 1) / TPB);      // 12500
  dim3 blkElem((SBIG + TPB - 1) / TPB);           // 50000
  dim3 blkGemm((MROWS / 16) * 4 / (TPB / 32));    // 6250 blocks, 8 waves each
  dim3 blkZero(4096);

  // ---- Layer 0 ----
  zero_f4<<<blkZero, TPB, 0, stream>>>((float4*)bufB, SBIG / 4);        // agg0 = 0
  zero_f4<<<1, TPB, 0, stream>>>((float4*)stats0, 512 / 4);             // stats+csum = 0
  add_pe<<<blkElem4, TPB, 0, stream>>>((const float4*)nf, (const float4*)pe, ids,
                                       (float4*)bufA);                  // h0
  scatter_edges<<<EE, TPB, 0, stream>>>(bufA, ew, esr, eds, bufB);      // agg0
  gemm_wmma_f32<<<blkGemm, TPB, 0, stream>>>(bufB, W0, bufA);           // x0 = agg0@W0
  bn_stats<<<1024, TPB, 0, stream>>>(bufA, stats0);
  bn_relu<<<blkElem, TPB, 0, stream>>>(bufA, stats0, g0, b0, bufB);     // h1

  // ---- Layer 1 ----
  zero_f4<<<blkZero, TPB, 0, stream>>>((float4*)bufA, SBIG / 4);        // agg1 = 0
  scatter_edges<<<EE, TPB, 0, stream>>>(bufB, ew, esr, eds, bufA);      // agg1
  col_sum<<<512, TPB, 0, stream>>>(bufA, csum);                         // sum over nodes
  readout_k<<<1, TPB, 0, stream>>>(csum, W1, out + SBIG);               // readout @ W1
  gemm_wmma_f32<<<blkGemm, TPB, 0, stream>>>(bufA, W1, bufB);           // x1 = agg1@W1
  bn_stats<<<1024, TPB, 0, stream>>>(bufB, stats1);
  bn_relu<<<blkElem, TPB, 0, stream>>>(bufB, stats1, g1, b1, out);      // h out
}